// AVWGCN_63565515981250
// MI455X (gfx1250) — compile-verified
//
#include <hip/hip_runtime.h>

typedef __attribute__((ext_vector_type(2))) float v2f;
typedef __attribute__((ext_vector_type(8))) float v8f;

#define N_NODES 2048
#define DIM 16
#define CIN 64
#define COUT 64
#define BATCH 64
#define KCH 3
#define WROWS (KCH * CIN)      // 192
#define WSTRIDE 72             // LDS row stride (floats), conflict-free

// ---------------------------------------------------------------------------
// Kernel 1: A = softmax(relu(E E^T), axis=1)   (one block per row)
// ---------------------------------------------------------------------------
__global__ void __launch_bounds__(256)
avw_support_kernel(const float* __restrict__ E, float* __restrict__ A) {
    __shared__ float row_e[DIM];
    __shared__ float vals[N_NODES];
    __shared__ float rbuf[256];
    const int tid = threadIdx.x;
    const int row = blockIdx.x;
    if (tid < DIM) row_e[tid] = E[row * DIM + tid];
    __syncthreads();

    float lmax = 0.0f;                       // relu >= 0 so 0 is a safe floor
    for (int c = tid; c < N_NODES; c += 256) {
        const float* ec = E + c * DIM;
        float dot = 0.f;
#pragma unroll
        for (int d = 0; d < DIM; ++d) dot += row_e[d] * ec[d];
        float v = dot > 0.f ? dot : 0.f;
        vals[c] = v;
        lmax = fmaxf(lmax, v);
    }
    rbuf[tid] = lmax;
    __syncthreads();
    for (int s = 128; s > 0; s >>= 1) {
        if (tid < s) rbuf[tid] = fmaxf(rbuf[tid], rbuf[tid + s]);
        __syncthreads();
    }
    const float m = rbuf[0];
    __syncthreads();

    float lsum = 0.f;
    for (int c = tid; c < N_NODES; c += 256) {
        float e = __expf(vals[c] - m);
        vals[c] = e;
        lsum += e;
    }
    rbuf[tid] = lsum;
    __syncthreads();
    for (int s = 128; s > 0; s >>= 1) {
        if (tid < s) rbuf[tid] += rbuf[tid + s];
        __syncthreads();
    }
    const float inv = 1.0f / rbuf[0];
    for (int c = tid; c < N_NODES; c += 256)
        A[(size_t)row * N_NODES + c] = vals[c] * inv;
}

// ---------------------------------------------------------------------------
// Kernel 2: Y[b] = alpha * A @ Xin[b] (+ beta * Xadd[b])
// One wave per (b, 16-row n-tile); 4 f32 WMMA accumulators cover all 64 cols.
// ---------------------------------------------------------------------------
__global__ void __launch_bounds__(128)
avw_spmm_kernel(const float* __restrict__ Ag, const float* __restrict__ Xin,
                const float* __restrict__ Xadd, float* __restrict__ Y,
                float alpha, float beta) {
    const int wave = threadIdx.x >> 5;
    const int lane = threadIdx.x & 31;
    const int half = lane >> 4;              // 0: K={0,1}, 1: K={2,3}
    const int l15  = lane & 15;
    const int flat = blockIdx.x * 4 + wave;  // 0 .. 8191
    const int b    = flat >> 7;              // batch
    const int n0   = (flat & 127) << 4;      // n-tile * 16

    // A fragment base: lane row = n0+l15, k-offset = half*2 (16x4 f32 layout)
    const float* Arow = Ag + (size_t)(n0 + l15) * N_NODES + half * 2;
    const float* Xb   = Xin + (size_t)b * N_NODES * CIN;

    v8f acc0 = {}, acc1 = {}, acc2 = {}, acc3 = {};
    for (int mm = 0; mm < N_NODES; mm += 4) {
        v2f a = *(const v2f*)(Arow + mm);
        // B fragment rows: K = mm + half*2 {+0,+1}; cols = cc*16 + l15
        const float* xr = Xb + (size_t)(mm + half * 2) * CIN + l15;
        v2f vb0, vb1, vb2, vb3;
        vb0[0] = xr[0];   vb0[1] = xr[CIN];
        vb1[0] = xr[16];  vb1[1] = xr[CIN + 16];
        vb2[0] = xr[32];  vb2[1] = xr[CIN + 32];
        vb3[0] = xr[48];  vb3[1] = xr[CIN + 48];
        acc0 = __builtin_amdgcn_wmma_f32_16x16x4_f32(false, a, false, vb0, (short)0, acc0, false, false);
        acc1 = __builtin_amdgcn_wmma_f32_16x16x4_f32(false, a, false, vb1, (short)0, acc1, false, false);
        acc2 = __builtin_amdgcn_wmma_f32_16x16x4_f32(false, a, false, vb2, (short)0, acc2, false, false);
        acc3 = __builtin_amdgcn_wmma_f32_16x16x4_f32(false, a, false, vb3, (short)0, acc3, false, false);
    }

    v8f accs[4] = {acc0, acc1, acc2, acc3};
#pragma unroll
    for (int cc = 0; cc < 4; ++cc) {
#pragma unroll
        for (int v = 0; v < 8; ++v) {
            const int rr  = n0 + v + half * 8;     // C/D layout: M = v + half*8
            const int col = cc * 16 + l15;
            const size_t o = ((size_t)b * N_NODES + rr) * CIN + col;
            float val = alpha * accs[cc][v];
            if (Xadd) val += beta * Xadd[o];
            Y[o] = val;
        }
    }
}

// ---------------------------------------------------------------------------
// Kernel 3: per-node fused weight generation + GEMM + bias
// out[:, n, :] = concat_k(x_g[k][:, n, :]) @ (E[n]·Wpool) + E[n]·bpool
// One block per node; W[n] (192x64) lives in LDS with stride-72 padding.
// ---------------------------------------------------------------------------
__global__ void __launch_bounds__(256)
avw_out_kernel(const float* __restrict__ xg0, const float* __restrict__ xg1,
               const float* __restrict__ xg2, const float* __restrict__ E,
               const float* __restrict__ Wp, const float* __restrict__ bp,
               float* __restrict__ out) {
    __shared__ float e[DIM];
    __shared__ float biasn[COUT];
    __shared__ float Wl[WROWS * WSTRIDE];    // ~55 KB
    const int tid = threadIdx.x;
    const int n = blockIdx.x;

    if (tid < DIM) e[tid] = E[n * DIM + tid];
    __syncthreads();

    if (tid < COUT) {
        float acc = 0.f;
#pragma unroll
        for (int d = 0; d < DIM; ++d) acc += e[d] * bp[d * COUT + tid];
        biasn[tid] = acc;
    }
    for (int j = tid; j < WROWS * COUT; j += 256) {
        float acc = 0.f;
#pragma unroll
        for (int d = 0; d < DIM; ++d)
            acc += e[d] * Wp[(size_t)d * (WROWS * COUT) + j];
        Wl[(j >> 6) * WSTRIDE + (j & 63)] = acc;
    }
    __syncthreads();

    const int wave = tid >> 5;
    const int lane = tid & 31;
    const int half = lane >> 4;
    const int l15  = lane & 15;
    const int b0   = (wave >> 1) * 16;       // tile pair shares b-rows
    const int c0   = (wave & 1) * 32;        // covers cols c0 and c0+16

    const float bA = biasn[c0 + l15];
    const float bB = biasn[c0 + 16 + l15];
    v8f acc0, acc1;
#pragma unroll
    for (int v = 0; v < 8; ++v) { acc0[v] = bA; acc1[v] = bB; }

#pragma unroll
    for (int k = 0; k < KCH; ++k) {
        const float* G = (k == 0) ? xg0 : (k == 1) ? xg1 : xg2;
        const float* Grow = G + ((size_t)(b0 + l15) * N_NODES + n) * CIN + half * 2;
        for (int i = 0; i < CIN; i += 4) {
            v2f a = *(const v2f*)(Grow + i);
            const int r = k * CIN + i + half * 2;
            v2f bf0, bf1;
            bf0[0] = Wl[r * WSTRIDE + c0 + l15];
            bf0[1] = Wl[(r + 1) * WSTRIDE + c0 + l15];
            bf1[0] = Wl[r * WSTRIDE + c0 + 16 + l15];
            bf1[1] = Wl[(r + 1) * WSTRIDE + c0 + 16 + l15];
            acc0 = __builtin_amdgcn_wmma_f32_16x16x4_f32(false, a, false, bf0, (short)0, acc0, false, false);
            acc1 = __builtin_amdgcn_wmma_f32_16x16x4_f32(false, a, false, bf1, (short)0, acc1, false, false);
        }
    }

#pragma unroll
    for (int v = 0; v < 8; ++v) {
        const int bb = b0 + v + half * 8;
        const size_t o = ((size_t)bb * N_NODES + n) * COUT;
        out[o + c0 + l15]      = acc0[v];
        out[o + c0 + 16 + l15] = acc1[v];
    }
}

// ---------------------------------------------------------------------------
extern "C" void kernel_launch(void* const* d_in, const int* in_sizes, int n_in,
                              void* d_out, int out_size, void* d_ws, size_t ws_size,
                              hipStream_t stream) {
    const float* x  = (const float*)d_in[0];   // [B, N, C_in]
    const float* E  = (const float*)d_in[1];   // [N, D]
    const float* Wp = (const float*)d_in[2];   // [D, K, C_in, C_out]
    const float* bp = (const float*)d_in[3];   // [D, C_out]
    float* out = (float*)d_out;                // [B, N, C_out]

    float* A  = (float*)d_ws;                              // N*N
    float* X1 = A  + (size_t)N_NODES * N_NODES;            // B*N*C
    float* X2 = X1 + (size_t)BATCH * N_NODES * CIN;        // B*N*C

    avw_support_kernel<<<N_NODES, 256, 0, stream>>>(E, A);

    const int spmm_blocks = (BATCH * (N_NODES / 16)) / 4;  // 4 waves/block
    // X1 = A @ x
    avw_spmm_kernel<<<spmm_blocks, 128, 0, stream>>>(A, x, nullptr, X1, 1.0f, 0.0f);
    // X2 = 2*A @ X1 - x   (T2 = 2*A*T1 - I applied to x)
    avw_spmm_kernel<<<spmm_blocks, 128, 0, stream>>>(A, X1, x, X2, 2.0f, -1.0f);

    avw_out_kernel<<<N_NODES, 256, 0, stream>>>(x, X1, X2, E, Wp, bp, out);
}